// DecoderSelfAttentionWithKcache_58016418234754
// MI455X (gfx1250) — compile-verified
//
#include <hip/hip_runtime.h>

typedef __attribute__((ext_vector_type(2))) float v2f;
typedef __attribute__((ext_vector_type(8))) float v8f;

#define D_MODEL 1024
#define NHEAD   16
#define HDIM    64
#define LALL    4096
#define BATCH   64

// ---------------------------------------------------------------------------
// C[64,1024] = A[64,1024] x op(B[1024,1024])
//   TRANSB=true : C[m,n] = sum_k A[m,k] * B[n,k]   (W.T projections)
//   TRANSB=false: C[m,n] = sum_k A[m,k] * B[k,n]   (@ W_kv)
// One wave32 per 16x16 tile; fp32 WMMA 16x16x4 accumulation (exact fp32).
// TRANSB is a template param so the 256-step inner loop is branch-free:
// load A pair, load B pair, v_wmma_f32_16x16x4_f32.
// VGPR layouts per CDNA5 ISA 7.12.2:
//   A 16x4 f32 : lanes 0-15 rows, half 0 -> K={0,1}, half 1 -> K={2,3}
//   B 4x16 f32 : lanes = N, VGPR0/1 hold K rows {0,1} (half 0) / {2,3} (half 1)
//   C/D 16x16  : VGPR r -> row r (lanes 0-15) / row r+8 (lanes 16-31)
// ---------------------------------------------------------------------------
template <bool TRANSB>
__global__ __launch_bounds__(32)
void gemm64_wmma(const float* __restrict__ A, const float* __restrict__ B,
                 float* __restrict__ C)
{
    const int tile = blockIdx.x;          // 256 tiles: 4 (M) x 64 (N)
    const int tm   = tile >> 6;
    const int tn   = tile & 63;
    const int lane = threadIdx.x & 31;
    const int half = lane >> 4;
    const int l    = lane & 15;

    const int m0 = tm * 16, n0 = tn * 16;
    v8f c = {};
    #pragma unroll 4
    for (int k = 0; k < D_MODEL; k += 4) {
        const int ka = k + half * 2;
        v2f a = *(const v2f*)(A + (m0 + l) * D_MODEL + ka);
        v2f b;
        if (TRANSB) {
            b = *(const v2f*)(B + (n0 + l) * D_MODEL + ka);
        } else {
            b.x = B[(ka + 0) * D_MODEL + n0 + l];
            b.y = B[(ka + 1) * D_MODEL + n0 + l];
        }
        c = __builtin_amdgcn_wmma_f32_16x16x4_f32(false, a, false, b,
                                                  (short)0, c, false, false);
    }
    #pragma unroll
    for (int r = 0; r < 8; ++r)
        C[(m0 + r + half * 8) * D_MODEL + n0 + l] = c[r];
}

// ---------------------------------------------------------------------------
// Flash-decode attention: one block per (b,h); 8 waves, each wave handles
// rows j = wave, wave+8, ...  Per row: one coalesced 256B non-temporal load
// of the 64-float K-slice (2 floats/lane), wave-wide butterfly dot, T5
// relative bucket bias + mask, online softmax; the per-lane k values feed
// the weighted accumulator directly (derived-V trick => K read exactly once).
// Single pass over the 1.07 GB K cache -> HBM-bound, ~46 us floor @ 23.3 TB/s.
// The appended K_new row (j = 4095) is handled by wave 7 after the main
// loop, so the streaming loop is branch-free.
// ---------------------------------------------------------------------------
struct OnlineState {
    float m, s, acc0, acc1;
};

__device__ __forceinline__
void online_update(OnlineState& st, int j, float k0, float k1,
                   float q0, float q1, int b, int h,
                   const float* __restrict__ bias_table,
                   const unsigned char* __restrict__ mask)
{
    float p = q0 * k0 + q1 * k1;
    #pragma unroll
    for (int off = 16; off; off >>= 1)
        p += __shfl_xor(p, off, 32);

    // relative-position bucket (unidirectional): rp = (Lall-1) - j >= 0
    const int rp = (LALL - 1) - j;
    int bucket;
    if (rp < 16) {                       // uniform across the wave
        bucket = rp;
    } else {
        const float inv_log8 = 0.48089834696298780245f; // 1/ln(8)
        int v = 16 + (int)(__logf((float)rp * 0.0625f) * (inv_log8 * 16.0f));
        bucket = (v < 31) ? v : 31;
    }
    float score = p * 0.125f + bias_table[bucket * NHEAD + h];
    score = mask[b * LALL + j] ? score : -10000.0f;

    const float mnew = fmaxf(st.m, score);
    const float corr = __expf(st.m - mnew);
    const float w    = __expf(score - mnew);
    st.s    = st.s * corr + w;
    st.acc0 = st.acc0 * corr + w * k0;
    st.acc1 = st.acc1 * corr + w * k1;
    st.m    = mnew;
}

__global__ __launch_bounds__(256)
void attn_decode(const float* __restrict__ Qn,        // [64,1024]
                 const float* __restrict__ Kn,        // [64,1024] (new K row)
                 const float* __restrict__ kcache,    // [64,4095,1024]
                 const float* __restrict__ bias_table,// [32,16]
                 const unsigned char* __restrict__ mask, // [64,4096]
                 float* __restrict__ ctx1)            // [64,1024]
{
    const int bh   = blockIdx.x;
    const int b    = bh >> 4;
    const int h    = bh & 15;
    const int tid  = threadIdx.x;
    const int wave = tid >> 5;
    const int lane = tid & 31;

    __shared__ float q_sh[HDIM];
    __shared__ float accsh[8][HDIM];
    __shared__ float msh[8];
    __shared__ float ssh[8];

    if (tid < HDIM) q_sh[tid] = Qn[b * D_MODEL + h * HDIM + tid];
    __syncthreads();

    const float q0 = q_sh[lane * 2];
    const float q1 = q_sh[lane * 2 + 1];
    const float* base = kcache + (size_t)b * (LALL - 1) * D_MODEL
                      + h * HDIM + lane * 2;

    OnlineState st = { -1e30f, 0.0f, 0.0f, 0.0f };

    // branch-free stream over the 4095 cached rows
    #pragma unroll 2
    for (int j = wave; j < LALL - 1; j += 8) {
        const float* krow = base + (size_t)j * D_MODEL;
        // prefetch ~8 wave-iterations ahead to hide HBM latency (read-once)
        __builtin_prefetch(krow + 64 * D_MODEL, 0, 0);
        v2f kv = __builtin_nontemporal_load((const v2f*)krow);
        online_update(st, j, kv.x, kv.y, q0, q1, b, h, bias_table, mask);
    }
    // the single appended K_new row (j = 4095 = 7 mod 8) -> wave 7
    if (wave == 7) {
        v2f kv = *(const v2f*)(Kn + b * D_MODEL + h * HDIM + lane * 2);
        online_update(st, LALL - 1, kv.x, kv.y, q0, q1, b, h, bias_table, mask);
    }

    accsh[wave][lane * 2]     = st.acc0;
    accsh[wave][lane * 2 + 1] = st.acc1;
    if (lane == 0) { msh[wave] = st.m; ssh[wave] = st.s; }
    __syncthreads();

    if (tid < HDIM) {
        float M = -1e30f;
        #pragma unroll
        for (int w = 0; w < 8; ++w) M = fmaxf(M, msh[w]);
        float S = 0.0f, A = 0.0f;
        #pragma unroll
        for (int w = 0; w < 8; ++w) {
            const float c = __expf(msh[w] - M);
            S += ssh[w] * c;
            A += accsh[w][tid] * c;
        }
        ctx1[b * D_MODEL + h * HDIM + tid] = A / S;
    }
}

// ---------------------------------------------------------------------------
extern "C" void kernel_launch(void* const* d_in, const int* in_sizes, int n_in,
                              void* d_out, int out_size, void* d_ws, size_t ws_size,
                              hipStream_t stream)
{
    (void)in_sizes; (void)n_in; (void)out_size; (void)ws_size;
    const float* hidden      = (const float*)d_in[0];  // [64,1,1024]
    const float* kcache      = (const float*)d_in[1];  // [64,4095,1024]
    const float* W_q         = (const float*)d_in[2];  // [1024,1024]
    const float* W_k         = (const float*)d_in[3];
    const float* W_kv        = (const float*)d_in[4];
    const float* W_o         = (const float*)d_in[5];
    const float* bias_table  = (const float*)d_in[6];  // [32,16]
    const unsigned char* msk = (const unsigned char*)d_in[7]; // [64,1,1,4096] bool
    float* out = (float*)d_out;                        // [64,1,1024]

    float* ws   = (float*)d_ws;
    float* Qn   = ws;            // 64*1024
    float* Kn   = ws + 65536;    // 64*1024
    float* ctx1 = ws + 131072;   // 64*1024
    float* tmp  = ws + 196608;   // 64*1024

    // Q/K projections (W.T -> fast-axis contraction: TRANSB=true)
    gemm64_wmma<true><<<256, 32, 0, stream>>>(hidden, W_q, Qn);
    gemm64_wmma<true><<<256, 32, 0, stream>>>(hidden, W_k, Kn);

    // Fused single-pass attention over the K cache (bandwidth-bound)
    attn_decode<<<BATCH * NHEAD, 256, 0, stream>>>(Qn, Kn, kcache, bias_table,
                                                   msk, ctx1);

    // context = (ctx1 @ W_kv) @ W_o.T
    gemm64_wmma<false><<<256, 32, 0, stream>>>(ctx1, W_kv, tmp);
    gemm64_wmma<true ><<<256, 32, 0, stream>>>(tmp, W_o, out);
}